// SlidingKernelAttention_12893491823372
// MI455X (gfx1250) — compile-verified
//
#include <hip/hip_runtime.h>
#include <math.h>
#include <stdint.h>

typedef float    v2f __attribute__((ext_vector_type(2)));
typedef float    v8f __attribute__((ext_vector_type(8)));
typedef uint32_t v4u __attribute__((ext_vector_type(4)));
typedef uint32_t v8u __attribute__((ext_vector_type(8)));

#define HW      67
#define HWHW    (67 * 67)
#define SEQ     256
#define SCALE_F 0.70710678118654752f   // (DIM/HEADS)^-0.5 = 2^-0.5

// ---- 16-lane butterfly reductions via DPP (no LDS traffic) ----
template <int CTRL>
__device__ __forceinline__ float dpp_src(float v) {
    return __int_as_float(
        __builtin_amdgcn_mov_dpp(__float_as_int(v), CTRL, 0xf, 0xf, true));
}
__device__ __forceinline__ float row16_max(float v) {
    v = fmaxf(v, dpp_src<0xB1>(v));    // quad_perm(1,0,3,2)  : xor1
    v = fmaxf(v, dpp_src<0x4E>(v));    // quad_perm(2,3,0,1)  : xor2
    v = fmaxf(v, dpp_src<0x141>(v));   // row_half_mirror     : pairs quads in 8
    v = fmaxf(v, dpp_src<0x140>(v));   // row_mirror          : pairs 8s in 16
    return v;
}
__device__ __forceinline__ float row16_sum(float v) {
    v += dpp_src<0xB1>(v);
    v += dpp_src<0x4E>(v);
    v += dpp_src<0x141>(v);
    v += dpp_src<0x140>(v);
    return v;
}

__device__ __forceinline__ v8f wmma4(v2f a, v2f b, v8f c) {
    return __builtin_amdgcn_wmma_f32_16x16x4_f32(false, a, false, b,
                                                 (short)0, c, false, false);
}

// One block per attention problem n (n = (b*C+c)*16 + i*4 + j).
// 256 threads = 8 waves; wave w owns query row-tiles {2w, 2w+1}.
__global__ __launch_bounds__(256) void ska_attn_kernel(const float* __restrict__ x,
                                                       const float* __restrict__ w_qkv,
                                                       float* __restrict__ out)
{
    __shared__ float sQ[SEQ][17];      // xu staging (TDM target), then Q rows
    __shared__ float sK[SEQ][17];      // K rows
    __shared__ float sV[SEQ][17];      // V rows
    __shared__ float sP[8][16][17];    // per-wave P-tile relayout scratch
    __shared__ float sW[48 * 17];      // w_qkv padded to stride 17

    const int t  = threadIdx.x;
    const int n  = blockIdx.x;
    const int bc = n >> 4;
    const int ij = n & 15;
    const int oi = ij >> 2;
    const int oj = ij & 3;

    const int lane = t & 31;
    const int wv   = t >> 5;
    const int m    = lane & 15;   // A row / B col / D col index
    const int g    = lane >> 4;   // K-group in fragments; D row offset = 8*g

    // ---- TDM: DMA the 64x64 fp32 window at (oi,oj) straight into sQ ----
    // LDS padding: pad_interval=3 (insert pad every 16 dwords), pad_amount=0
    // (1 dword) reproduces the [256][17] token layout exactly: image row y
    // lands in tokens 4y..4y+3 (16 payload floats + 1 pad each).
    if (wv == 0) {
        uint64_t gaddr = (uint64_t)(uintptr_t)(x + (size_t)bc * HWHW + oi * HW + oj);
        uint32_t ldsa  = (uint32_t)(uintptr_t)&sQ[0][0];   // AS3 offset in low 32b

        v4u g0;
        g0[0] = 1u;                                   // count=1, user descriptor
        g0[1] = ldsa;                                 // lds_addr
        g0[2] = (uint32_t)gaddr;                      // global_addr[31:0]
        g0[3] = (uint32_t)(gaddr >> 32) | 0x80000000u;// global_addr[56:32] | type=2

        v8u g1;
        g1[0] = (2u << 16)      // data_size = 4 bytes
              | (1u << 20)      // pad_enable
              | (3u << 22);     // pad_interval = 16 dwords (pad_amount=0 -> 1 dword)
        g1[1] = (67u << 16);    // tensor_dim0 = 67 (bits 79:48)
        g1[2] = (67u << 16);    // tensor_dim1 = 67 (bits 111:80)
        g1[3] = (64u << 16);    // tile_dim0 = 64  (bits 127:112)
        g1[4] = 64u;            // tile_dim1 = 64, tile_dim2 = 0
        g1[5] = 67u;            // tensor_dim0_stride = 67 (bits 207:160)
        g1[6] = 0u;             // tensor_dim1_stride (unused for 2D)
        g1[7] = 0u;

        asm volatile("tensor_load_to_lds %0, %1"
                     :
                     : "s"(g0), "s"(g1)
                     : "memory");
        __builtin_amdgcn_s_wait_tensorcnt(0);
    }

    // ---- stage weights (padded) ----
    #pragma unroll
    for (int r = 0; r < 3; ++r) {
        int f = t + 256 * r;
        sW[(f >> 4) * 17 + (f & 15)] = w_qkv[f];
    }
    __syncthreads();   // xu (via TDM) + W staged

    // ---- W^T B-fragments, shared across both owned row-tiles ----
    v2f wbq[4], wbk[4], wbv[4];
    #pragma unroll
    for (int c = 0; c < 4; ++c) {
        int col = 4 * c + 2 * g;
        wbq[c][0] = sW[ m        * 17 + col];  wbq[c][1] = sW[ m        * 17 + col + 1];
        wbk[c][0] = sW[(m + 16) * 17 + col];   wbk[c][1] = sW[(m + 16) * 17 + col + 1];
        wbv[c][0] = sW[(m + 32) * 17 + col];   wbv[c][1] = sW[(m + 32) * 17 + col + 1];
    }

    // ---- QKV via WMMA: D(Q/K/V tile) = xu_tile(16x16) @ W*^T(16x16) ----
    #pragma unroll
    for (int rtl = 0; rtl < 2; ++rtl) {
        const int rt = wv * 2 + rtl;
        v2f xa[4];
        #pragma unroll
        for (int c = 0; c < 4; ++c) {
            xa[c][0] = sQ[rt * 16 + m][4 * c + 2 * g];
            xa[c][1] = sQ[rt * 16 + m][4 * c + 2 * g + 1];
        }
        v8f Qd, Kd, Vd;
        #pragma unroll
        for (int r = 0; r < 8; ++r) { Qd[r] = 0.f; Kd[r] = 0.f; Vd[r] = 0.f; }
        #pragma unroll
        for (int c = 0; c < 4; ++c) {
            Qd = wmma4(xa[c], wbq[c], Qd);
            Kd = wmma4(xa[c], wbk[c], Kd);
            Vd = wmma4(xa[c], wbv[c], Vd);
        }
        // D-layout scatter back to row buffers (xu rows already consumed)
        #pragma unroll
        for (int r = 0; r < 8; ++r) {
            sQ[rt * 16 + r + 8 * g][m] = Qd[r];
            sK[rt * 16 + r + 8 * g][m] = Kd[r];
            sV[rt * 16 + r + 8 * g][m] = Vd[r];
        }
    }
    __syncthreads();   // Q/K/V ready for all waves

    for (int rtl = 0; rtl < 2; ++rtl) {
        const int rt = wv * 2 + rtl;

        // Q A-fragments (reused across all 16 col-tiles)
        v2f qa[4];
        #pragma unroll
        for (int c = 0; c < 4; ++c) {
            qa[c][0] = sQ[rt * 16 + m][4 * c + 2 * g];
            qa[c][1] = sQ[rt * 16 + m][4 * c + 2 * g + 1];
        }

        // ---- pass 1: full score row for this row-tile, kept in registers ----
        v8f S[16];                       // 128 VGPRs/lane
        #pragma unroll
        for (int ct = 0; ct < 16; ++ct) {
            v8f acc;
            #pragma unroll
            for (int r = 0; r < 8; ++r) acc[r] = 0.f;
            #pragma unroll
            for (int c = 0; c < 4; ++c) {
                v2f kb;   // B = K^T chunk: B[kf][col] = K[ct*16+col][4c+kf]
                kb[0] = sK[ct * 16 + m][4 * c + 2 * g];
                kb[1] = sK[ct * 16 + m][4 * c + 2 * g + 1];
                acc = wmma4(qa[c], kb, acc);
            }
            S[ct] = acc;
        }

        // ---- pass 2: exact softmax (lane-local fold + 16-lane DPP butterfly) ----
        float mrow[8], lsum[8];
        #pragma unroll
        for (int r = 0; r < 8; ++r) {
            float mx = S[0][r];
            #pragma unroll
            for (int ct = 1; ct < 16; ++ct) mx = fmaxf(mx, S[ct][r]);
            mrow[r] = row16_max(mx) * SCALE_F;
            lsum[r] = 0.f;
        }
        #pragma unroll
        for (int ct = 0; ct < 16; ++ct) {
            #pragma unroll
            for (int r = 0; r < 8; ++r) {
                float p = __expf(S[ct][r] * SCALE_F - mrow[r]);
                S[ct][r] = p;
                lsum[r] += p;
            }
        }
        #pragma unroll
        for (int r = 0; r < 8; ++r) lsum[r] = row16_sum(lsum[r]);

        // ---- pass 3: O = P @ V (relayout each P tile via per-wave LDS bounce) ----
        v8f O;
        #pragma unroll
        for (int r = 0; r < 8; ++r) O[r] = 0.f;

        #pragma unroll
        for (int ct = 0; ct < 16; ++ct) {
            // V B-fragments first: independent of the sP bounce
            v2f vb[4];
            #pragma unroll
            for (int c = 0; c < 4; ++c) {
                vb[c][0] = sV[ct * 16 + 4 * c + 2 * g][m];
                vb[c][1] = sV[ct * 16 + 4 * c + 2 * g + 1][m];
            }
            asm volatile("" ::: "memory");
            #pragma unroll
            for (int r = 0; r < 8; ++r)
                sP[wv][r + 8 * g][m] = S[ct][r];           // D-layout scatter
            asm volatile("s_wait_dscnt 0x0" ::: "memory"); // stores visible to wave
            v2f pa[4];
            #pragma unroll
            for (int c = 0; c < 4; ++c) {
                pa[c][0] = sP[wv][m][4 * c + 2 * g];       // A-layout gather
                pa[c][1] = sP[wv][m][4 * c + 2 * g + 1];
            }
            #pragma unroll
            for (int c = 0; c < 4; ++c)
                O = wmma4(pa[c], vb[c], O);
            asm volatile("" ::: "memory");  // keep next iter's stores after these loads
        }

        // ---- normalize + overlap-add fold ----
        #pragma unroll
        for (int r = 0; r < 8; ++r) {
            float val  = O[r] / lsum[r];
            int   srow = rt * 16 + r + 8 * g;   // sequence index
            int   l    = srow * 16 + m;         // flat pos in 64x64 patch grid
            int   phh  = l >> 6;
            int   pww  = l & 63;
            atomicAdd(out + (size_t)bc * HWHW + (phh + oi) * HW + (pww + oj), val);
        }
    }
}

extern "C" void kernel_launch(void* const* d_in, const int* in_sizes, int n_in,
                              void* d_out, int out_size, void* d_ws, size_t ws_size,
                              hipStream_t stream)
{
    const float* x = (const float*)d_in[0];   // [2,64,67,67] fp32
    const float* w = (const float*)d_in[1];   // [48,16] fp32
    float* out = (float*)d_out;               // [2,64,67,67] fp32

    // output is overlap-add accumulated -> must start from zero every call
    hipMemsetAsync(out, 0, (size_t)out_size * sizeof(float), stream);

    ska_attn_kernel<<<2048, 256, 0, stream>>>(x, w, out);
}